// STGCN_6219112645312
// MI455X (gfx1250) — compile-verified
//
#include <hip/hip_runtime.h>
#include <cstdint>

// ---------------------------------------------------------------------------
// STGCN on MI455X (gfx1250): all dense stages as f32 WMMA 16x16x4 GEMMs.
// K=3 temporal convs are decomposed into 3 shifted GEMMs over a zero-padded
// LDS activation tile (no im2col build, no divisions in hot loops).
// Graph scatter via per-wave f32 atomics (lands in 192MB L2) + prefetch.
// ---------------------------------------------------------------------------

typedef float v2f __attribute__((ext_vector_type(2)));
typedef float v8f __attribute__((ext_vector_type(8)));

__device__ __forceinline__ v8f wmma4(v2f a, v2f b, v8f c) {
  // D = A(16x4 f32) * B(4x16 f32) + C(16x16 f32)
  return __builtin_amdgcn_wmma_f32_16x16x4_f32(
      /*neg_a=*/false, a, /*neg_b=*/false, b,
      /*c_mod=*/(short)0, c, /*reuse_a=*/false, /*reuse_b=*/false);
}

// ------------------------------- utilities ---------------------------------

__global__ void fill_kernel(float* __restrict__ p, float v, size_t n) {
  size_t i = (size_t)blockIdx.x * blockDim.x + threadIdx.x;
  if (i < n) p[i] = v;
}

__global__ void deg_kernel(const int* __restrict__ dst, const float* __restrict__ ew,
                           float* __restrict__ deg, int E) {
  int e = blockIdx.x * blockDim.x + threadIdx.x;
  if (e < E) atomicAdd(&deg[dst[e]], ew[e]);
}

__global__ void dis_kernel(const float* __restrict__ deg, float* __restrict__ dis, int n) {
  int i = blockIdx.x * blockDim.x + threadIdx.x;
  if (i < n) {
    float d = deg[i];
    dis[i] = (d > 0.f) ? rsqrtf(d) : 0.f;
  }
}

// ------------------- gated temporal conv (3 shifted GEMMs) -----------------
// x: (N, Cin, 12), w: (128, Cin, 3), b: (128), out: (N, 64, 12)
// Block: 128 threads = 4 waves; 4 nodes -> 48 columns = 3 col tiles.
// LDS tile ldsX[c][nl*14 + 1 + t], zero pads at t'=0 and t'=13 absorb the
// conv boundary, so the k-loop B reads need no masking at all.
// Wave w computes row-tile w (P channels) and w+4 (Q channels): gate in regs.
__global__ void __launch_bounds__(128)
gated_tconv_wmma(const float* __restrict__ x, const float* __restrict__ w,
                 const float* __restrict__ bias, float* __restrict__ out,
                 int Cin, int log2Cin, int N) {
  __shared__ float ldsX[64 * 56];  // [c][col'] col' = nl*14 + 1 + t (max 14.3KB)
  const int KTOT = Cin * 3;
  const int node0 = blockIdx.x * 4;
  const int tid = threadIdx.x;

  // Zero the 2 pad slots per (c, node).
  for (int j = tid; j < Cin * 8; j += 128) {
    int c = j >> 3, r = j & 7;
    int nl = r >> 1, side = r & 1;
    ldsX[c * 56 + nl * 14 + side * 13] = 0.f;
  }
  // Copy x tile: 4 nodes are 4*Cin*12 contiguous floats. Shift/mask indexing.
  const float* xs = x + (size_t)node0 * Cin * 12;
  const int total = Cin * 48;
  for (int i = tid; i < total; i += 128) {
    int q = i / 12;            // single cheap const-division
    int t = i - q * 12;
    int c = q & (Cin - 1);
    int nl = q >> log2Cin;
    float v = (node0 + nl < N) ? xs[i] : 0.f;
    ldsX[c * 56 + nl * 14 + 1 + t] = v;
  }
  __syncthreads();

  const int lane = tid & 31;
  const int wid = tid >> 5;
  const int m = lane & 15;      // A row within tile / B,D column within tile
  const int khalf = lane >> 4;  // which K pair this half-wave holds

  // Per-column bases (computed once): nl*14 + 1 + t, plus (nl, t) for stores.
  int cbase[3], nlA[3], tA[3];
#pragma unroll
  for (int ct = 0; ct < 3; ++ct) {
    int col = ct * 16 + m;
    int nl = col / 12;
    int t = col - nl * 12;
    nlA[ct] = nl;
    tA[ct] = t;
    cbase[ct] = nl * 14 + 1 + t;
  }

  v8f accP[3] = {};
  v8f accQ[3] = {};
  const float* wP = w + (size_t)(wid * 16 + m) * KTOT;        // P channels
  const float* wQ = w + (size_t)(wid * 16 + 64 + m) * KTOT;   // Q channels

  for (int dk = 0; dk < 3; ++dk) {
    const float* wPd = wP + dk;
    const float* wQd = wQ + dk;
    int cb0 = cbase[0] + dk - 1;
    int cb1 = cbase[1] + dk - 1;
    int cb2 = cbase[2] + dk - 1;
    for (int kk = 0; kk < Cin; kk += 4) {
      const int kA = kk + khalf * 2;  // input-channel pair base
      v2f aP; aP.x = wPd[kA * 3]; aP.y = wPd[kA * 3 + 3];
      v2f aQ; aQ.x = wQd[kA * 3]; aQ.y = wQd[kA * 3 + 3];
      const float* r0 = &ldsX[kA * 56];
      const float* r1 = r0 + 56;
      v2f b0; b0.x = r0[cb0]; b0.y = r1[cb0];
      v2f b1; b1.x = r0[cb1]; b1.y = r1[cb1];
      v2f b2; b2.x = r0[cb2]; b2.y = r1[cb2];
      accP[0] = wmma4(aP, b0, accP[0]);
      accQ[0] = wmma4(aQ, b0, accQ[0]);
      accP[1] = wmma4(aP, b1, accP[1]);
      accQ[1] = wmma4(aQ, b1, accQ[1]);
      accP[2] = wmma4(aP, b2, accP[2]);
      accQ[2] = wmma4(aQ, b2, accQ[2]);
    }
  }

  // Gate P * sigmoid(Q) and store.
#pragma unroll
  for (int ct = 0; ct < 3; ++ct) {
    int n = node0 + nlA[ct];
    if (n >= N) continue;
    int t = tA[ct];
#pragma unroll
    for (int v = 0; v < 8; ++v) {
      int ch = wid * 16 + khalf * 8 + v;  // output channel (0..63)
      float P = accP[ct][v] + bias[ch];
      float Q = accQ[ct][v] + bias[ch + 64];
      float s = 1.f / (1.f + __expf(-Q));
      out[((size_t)n * 64 + ch) * 12 + t] = P * s;
    }
  }
}

// ----------------------- GCN pointwise: xt = W @ h -------------------------
// h: (N,64,12), W: (64,64), xt: (N,64,12). Same tiling, K=64, no gate/bias.
__global__ void __launch_bounds__(128)
pointwise_wmma(const float* __restrict__ h, const float* __restrict__ W,
               float* __restrict__ xt, int N) {
  const int COLS = 48;
  __shared__ float ldsB[64 * 48];  // [c][col]
  const int node0 = blockIdx.x * 4;
  const int tid = threadIdx.x;

  const float* hs = h + (size_t)node0 * 768;
  for (int i = tid; i < 4 * 768; i += 128) {
    int q = i / 12;            // q = nl*64 + c
    int t = i - q * 12;
    int c = q & 63;
    int nl = q >> 6;
    float v = (node0 + nl < N) ? hs[i] : 0.f;
    ldsB[c * COLS + nl * 12 + t] = v;
  }
  __syncthreads();

  const int lane = tid & 31;
  const int wid = tid >> 5;
  const int m = lane & 15;
  const int khalf = lane >> 4;

  int nlA[3], tA[3];
#pragma unroll
  for (int ct = 0; ct < 3; ++ct) {
    int col = ct * 16 + m;
    nlA[ct] = col / 12;
    tA[ct] = col - nlA[ct] * 12;
  }

  v8f acc[3] = {};
  const float* wR = W + (size_t)(wid * 16 + m) * 64;

  for (int kk = 0; kk < 64; kk += 4) {
    const int kA = kk + khalf * 2;
    v2f a; a.x = wR[kA]; a.y = wR[kA + 1];
    const float* b0 = &ldsB[(size_t)kA * COLS + m];
    const float* b1 = &ldsB[(size_t)(kA + 1) * COLS + m];
#pragma unroll
    for (int ct = 0; ct < 3; ++ct) {
      v2f bb; bb.x = b0[ct * 16]; bb.y = b1[ct * 16];
      acc[ct] = wmma4(a, bb, acc[ct]);
    }
  }

#pragma unroll
  for (int ct = 0; ct < 3; ++ct) {
    int n = node0 + nlA[ct];
    if (n >= N) continue;
#pragma unroll
    for (int v = 0; v < 8; ++v) {
      int ch = wid * 16 + khalf * 8 + v;
      xt[((size_t)n * 64 + ch) * 12 + tA[ct]] = acc[ct][v];
    }
  }
}

// --------------------------- GCN edge scatter ------------------------------
// acc[dst] += xt[src] * (dis[src]*ew*dis[dst]); one wave per edge.
// Prefetch the 3KB source row (one byte per 128B line per lane) while the
// normalization coefficient loads are in flight.
__global__ void scatter_kernel(const float* __restrict__ xt, const int* __restrict__ src,
                               const int* __restrict__ dst, const float* __restrict__ ew,
                               const float* __restrict__ dis, float* __restrict__ acc, int E) {
  int gw = (int)((blockIdx.x * blockDim.x + threadIdx.x) >> 5);
  int lane = threadIdx.x & 31;
  if (gw >= E) return;
  int s = src[gw], d = dst[gw];
  const float* xs = xt + (size_t)s * 768;
  float* ad = acc + (size_t)d * 768;
  __builtin_prefetch(xs + (size_t)lane * 32, 0, 0);  // global_prefetch_b8
  float coef = dis[s] * ew[gw] * dis[d];
#pragma unroll
  for (int i = 0; i < 24; ++i)
    atomicAdd(ad + lane + i * 32, xs[lane + i * 32] * coef);
}

// Self-loop contribution + bias + ReLU (in place on acc). One block per node:
// the /768 lives on the SALU via blockIdx, only a cheap /12 remains.
__global__ void gcn_finish_kernel(float* __restrict__ acc, const float* __restrict__ xt,
                                  const float* __restrict__ dis,
                                  const float* __restrict__ bias) {
  const int n = blockIdx.x;
  const float ds = dis[n];
  const float d2 = ds * ds;
  const size_t base = (size_t)n * 768;
  for (int r = threadIdx.x; r < 768; r += 256) {
    int c = r / 12;
    float v = acc[base + r] + xt[base + r] * d2 + bias[c];
    acc[base + r] = v > 0.f ? v : 0.f;
  }
}

// ------------------------ final conv: (12 x 768) GEMM ----------------------
// h: (N,64,12) flattened K=768 per node; fw: (12,768); out: (N,12).
// Block = 4 waves; each wave owns 16 nodes (B columns); fw zero-padded to 16
// rows in LDS with a division-free row-sliced fill.
__global__ void __launch_bounds__(128)
fin_wmma(const float* __restrict__ h, const float* __restrict__ fw,
         const float* __restrict__ fb, float* __restrict__ out, int N) {
  __shared__ float ldsA[16 * 768];  // 49.2KB
  const int tid = threadIdx.x;
  {
    const int row = tid >> 3;        // 16 rows, 8 threads each
    const int k0 = (tid & 7) * 96;   // 8 chunks of 96
    const bool vr = row < 12;
    const float* srcRow = fw + (size_t)(vr ? row : 0) * 768;
    float* dstRow = ldsA + (size_t)row * 768 + k0;
    for (int j = 0; j < 96; ++j) dstRow[j] = vr ? srcRow[k0 + j] : 0.f;
  }
  __syncthreads();

  const int lane = tid & 31;
  const int wid = tid >> 5;
  const int m = lane & 15;
  const int khalf = lane >> 4;

  const int node = blockIdx.x * 64 + wid * 16 + m;
  const bool valid = node < N;
  const float* hb = h + (size_t)(valid ? node : 0) * 768;
  __builtin_prefetch(hb + (size_t)khalf * 384, 0, 0);

  v8f acc = {};
  const float* aRow = ldsA + (size_t)m * 768;
  for (int kk = 0; kk < 768; kk += 4) {
    const int kA = kk + khalf * 2;
    v2f a; a.x = aRow[kA]; a.y = aRow[kA + 1];
    v2f bb; bb.x = hb[kA]; bb.y = hb[kA + 1];
    acc = wmma4(a, bb, acc);
  }

  if (!valid) return;
#pragma unroll
  for (int v = 0; v < 8; ++v) {
    int row = khalf * 8 + v;
    if (row < 12) out[(size_t)node * 12 + row] = acc[v] + fb[row];
  }
}

// ------------------------------- launcher ----------------------------------

extern "C" void kernel_launch(void* const* d_in, const int* in_sizes, int n_in,
                              void* d_out, int out_size, void* d_ws, size_t ws_size,
                              hipStream_t stream) {
  const float* x      = (const float*)d_in[0];
  const int*   eidx   = (const int*)d_in[1];
  const float* ew     = (const float*)d_in[2];
  const float* tc1a_w = (const float*)d_in[3];
  const float* tc1a_b = (const float*)d_in[4];
  const float* gc1_w  = (const float*)d_in[5];
  const float* gc1_b  = (const float*)d_in[6];
  const float* tc1b_w = (const float*)d_in[7];
  const float* tc1b_b = (const float*)d_in[8];
  const float* tc2a_w = (const float*)d_in[9];
  const float* tc2a_b = (const float*)d_in[10];
  const float* gc2_w  = (const float*)d_in[11];
  const float* gc2_b  = (const float*)d_in[12];
  const float* tc2b_w = (const float*)d_in[13];
  const float* tc2b_b = (const float*)d_in[14];
  const float* fin_w  = (const float*)d_in[15];
  const float* fin_b  = (const float*)d_in[16];
  float* out = (float*)d_out;

  const int N = 10000, E = 160000;
  const int* src = eidx;
  const int* dst = eidx + E;

  const size_t bufElems = (size_t)N * 64 * 12;  // 7.68M floats = 30.7MB
  float* buf0 = (float*)d_ws;
  float* buf1 = buf0 + bufElems;
  float* buf2 = buf1 + bufElems;
  float* deg  = buf2 + bufElems;
  float* dis  = deg + N;

  const int convBlocks = (N + 3) / 4;                      // 2500
  const int zeroBlocks = (int)((bufElems + 255) / 256);    // 30000
  const int edgeBlocks = (E * 32 + 255) / 256;             // 20000

  // Degree normalization (self loops contribute 1.0 each).
  fill_kernel<<<(N + 255) / 256, 256, 0, stream>>>(deg, 1.0f, (size_t)N);
  deg_kernel<<<(E + 255) / 256, 256, 0, stream>>>(dst, ew, deg, E);
  dis_kernel<<<(N + 255) / 256, 256, 0, stream>>>(deg, dis, N);

  // Layer 1: gated tconv (Cin=32) -> GCN -> gated tconv (Cin=64)
  gated_tconv_wmma<<<convBlocks, 128, 0, stream>>>(x, tc1a_w, tc1a_b, buf0, 32, 5, N);
  pointwise_wmma<<<convBlocks, 128, 0, stream>>>(buf0, gc1_w, buf1, N);
  fill_kernel<<<zeroBlocks, 256, 0, stream>>>(buf2, 0.0f, bufElems);
  scatter_kernel<<<edgeBlocks, 256, 0, stream>>>(buf1, src, dst, ew, dis, buf2, E);
  gcn_finish_kernel<<<N, 256, 0, stream>>>(buf2, buf1, dis, gc1_b);
  gated_tconv_wmma<<<convBlocks, 128, 0, stream>>>(buf2, tc1b_w, tc1b_b, buf0, 64, 6, N);

  // Layer 2: gated tconv -> GCN -> gated tconv
  gated_tconv_wmma<<<convBlocks, 128, 0, stream>>>(buf0, tc2a_w, tc2a_b, buf1, 64, 6, N);
  pointwise_wmma<<<convBlocks, 128, 0, stream>>>(buf1, gc2_w, buf0, N);
  fill_kernel<<<zeroBlocks, 256, 0, stream>>>(buf2, 0.0f, bufElems);
  scatter_kernel<<<edgeBlocks, 256, 0, stream>>>(buf0, src, dst, ew, dis, buf2, E);
  gcn_finish_kernel<<<N, 256, 0, stream>>>(buf2, buf0, dis, gc2_b);
  gated_tconv_wmma<<<convBlocks, 128, 0, stream>>>(buf2, tc2b_w, tc2b_b, buf0, 64, 6, N);

  // Final projection to (N, 12).
  fin_wmma<<<(N + 63) / 64, 128, 0, stream>>>(buf0, fin_w, fin_b, out, N);
}